// COMSAGPool_b_89060441850423
// MI455X (gfx1250) — compile-verified
//
#include <hip/hip_runtime.h>
#include <math.h>

// Problem constants (match reference)
#define BGRAPH 64
#define NNODE  8192
#define DIM    128
#define KKEEP  4096            // N * 0.5
#define NTOT   (BGRAPH * NNODE)     // 524288
#define NEDGE  (BGRAPH * NNODE * 16) // 8388608

typedef __attribute__((ext_vector_type(2))) float v2f;
typedef __attribute__((ext_vector_type(8))) float v8f;

// ---------------------------------------------------------------- init
__global__ void k_init(unsigned* deg_out, unsigned* deg_in, float* agg, float* red) {
    int i = blockIdx.x * blockDim.x + threadIdx.x;
    if (i < NTOT) { deg_out[i] = 0u; deg_in[i] = 0u; agg[i] = 0.0f; }
    if (i == 0) { red[0] = -INFINITY; red[1] = 0.0f; }
}

// ---------------------------------------------------------------- degrees
__global__ void k_deg(const int* __restrict__ src, const int* __restrict__ dst,
                      unsigned* deg_out, unsigned* deg_in) {
    int e = blockIdx.x * blockDim.x + threadIdx.x;
    if (e < NEDGE) {
        atomicAdd(&deg_out[src[e]], 1u);
        atomicAdd(&deg_in[dst[e]], 1u);
    }
}

// ---------------------------------------------------------------- h = x @ W  (f32 WMMA)
// One wave computes h for 16 consecutive nodes.
// A (16x4 f32): lanes 0-15 hold M=0..15 with K={k,k+1}; lanes 16-31 hold M=0..15 with K={k+2,k+3}.
// B (4x16 f32): W[k..k+3] broadcast to all 16 columns -> every lane supplies W values for its
//               K positions; since all columns are identical, D[:,0] = x_rows @ W.
// D (16x16 f32, 8 VGPRs): column N=0 lives in lane 0 (M=0..7) and lane 16 (M=8..15).
__global__ void k_h_wmma(const float* __restrict__ x, const float* __restrict__ W,
                         float* __restrict__ h) {
    const int lane    = threadIdx.x & 31;
    const int wave    = (blockIdx.x * blockDim.x + threadIdx.x) >> 5;
    const int rowBase = wave * 16;
    const int m       = lane & 15;
    const int koff    = (lane >> 4) * 2;           // 0 for lanes 0-15, 2 for lanes 16-31
    const float* xrow = x + (size_t)(rowBase + m) * DIM;

    v8f c = {};
    for (int k = 0; k < DIM; k += 4) {
        v2f a = *(const v2f*)(xrow + k + koff);    // 8B aligned: k+koff is even
        v2f b;
        b.x = W[k + koff];
        b.y = W[k + koff + 1];
        c = __builtin_amdgcn_wmma_f32_16x16x4_f32(
                /*neg_a=*/false, a, /*neg_b=*/false, b,
                /*c_mod=*/(short)0, c, /*reuse_a=*/false, /*reuse_b=*/false);
    }
    if (lane == 0) {
        for (int r = 0; r < 8; ++r) h[rowBase + r] = c[r];
    } else if (lane == 16) {
        for (int r = 0; r < 8; ++r) h[rowBase + 8 + r] = c[r];
    }
}

// ---------------------------------------------------------------- edge aggregation
__global__ void k_edge(const int* __restrict__ src, const int* __restrict__ dst,
                       const float* __restrict__ h, const unsigned* __restrict__ deg_out,
                       float* agg) {
    int e = blockIdx.x * blockDim.x + threadIdx.x;
    if (e < NEDGE) {
        int s = src[e];
        unsigned d = deg_out[s];
        float nrm = rsqrtf((float)(d ? d : 1u));
        atomicAdd(&agg[dst[e]], h[s] * nrm);
    }
}

// ---------------------------------------------------------------- score + global max
__global__ void k_score(const float* __restrict__ agg, const unsigned* __restrict__ deg_in,
                        const float* __restrict__ bptr, float* score, float* red) {
    __shared__ float smax[256];
    int i = blockIdx.x * blockDim.x + threadIdx.x;
    unsigned d = deg_in[i];
    float s = agg[i] * rsqrtf((float)(d ? d : 1u)) + bptr[0];
    score[i] = s;
    smax[threadIdx.x] = s;
    __syncthreads();
    for (int off = 128; off > 0; off >>= 1) {
        if (threadIdx.x < off)
            smax[threadIdx.x] = fmaxf(smax[threadIdx.x], smax[threadIdx.x + off]);
        __syncthreads();
    }
    if (threadIdx.x == 0) atomicMax(&red[0], smax[0]);   // global_atomic_max_num_f32
}

// ---------------------------------------------------------------- sum of exp
__global__ void k_sumexp(const float* __restrict__ score, float* red) {
    __shared__ float ssum[256];
    int i = blockIdx.x * blockDim.x + threadIdx.x;
    float mx = red[0];
    ssum[threadIdx.x] = expf(score[i] - mx);
    __syncthreads();
    for (int off = 128; off > 0; off >>= 1) {
        if (threadIdx.x < off) ssum[threadIdx.x] += ssum[threadIdx.x + off];
        __syncthreads();
    }
    if (threadIdx.x == 0) atomicAdd(&red[1], ssum[0]);
}

// ---------------------------------------------------------------- softmax write
__global__ void k_soft(const float* __restrict__ score, const float* __restrict__ red,
                       float* out_soft) {
    int i = blockIdx.x * blockDim.x + threadIdx.x;
    out_soft[i] = expf(score[i] - red[0]) / red[1];
}

// ---------------------------------------------------------------- per-graph top-K
// One block (1024 threads) per graph; full bitonic sort of 8192 (key desc, idx asc),
// then mask + prefix scan for ascending complement. 64KB LDS.
__global__ __launch_bounds__(1024) void k_topk(const float* __restrict__ score,
                                               int* perm, int* perm_com) {
    __shared__ float s_key[NNODE];
    __shared__ int   s_idx[NNODE];
    const int g = blockIdx.x, t = threadIdx.x, base = g * NNODE;

    for (int i = t; i < NNODE; i += 1024) { s_key[i] = score[base + i]; s_idx[i] = i; }
    __syncthreads();

    for (int k = 2; k <= NNODE; k <<= 1) {
        for (int j = k >> 1; j > 0; j >>= 1) {
            for (int l = t; l < NNODE; l += 1024) {
                int p = l ^ j;
                if (p > l) {
                    bool dir = ((l & k) == 0);
                    float ka = s_key[l], kb = s_key[p];
                    int   ia = s_idx[l], ib = s_idx[p];
                    // "a after b" in (desc key, asc idx) order
                    bool gt = (ka < kb) || (ka == kb && ia > ib);
                    if (gt == dir) {
                        s_key[l] = kb; s_key[p] = ka;
                        s_idx[l] = ib; s_idx[p] = ia;
                    }
                }
            }
            __syncthreads();
        }
    }

    for (int j = t; j < KKEEP; j += 1024) perm[g * KKEEP + j] = base + s_idx[j];
    __syncthreads();

    int* mask = (int*)s_key;                      // scores no longer needed
    for (int i = t; i < NNODE; i += 1024) mask[i] = 1;
    __syncthreads();
    for (int j = t; j < KKEEP; j += 1024) mask[s_idx[j]] = 0;
    __syncthreads();

    // prefix scan over mask: each thread owns 8 consecutive entries
    const int lo = t * 8;
    int part = 0;
    for (int q = 0; q < 8; ++q) part += mask[lo + q];
    __syncthreads();                              // done reading s_idx
    s_idx[t] = part;
    __syncthreads();
    for (int off = 1; off < 1024; off <<= 1) {
        int v = s_idx[t];
        int add = (t >= off) ? s_idx[t - off] : 0;
        __syncthreads();
        s_idx[t] = v + add;
        __syncthreads();
    }
    int pos = s_idx[t] - part;                    // exclusive prefix
    for (int q = 0; q < 8; ++q) {
        if (mask[lo + q]) {
            perm_com[g * (NNODE - KKEEP) + pos] = base + lo + q;
            ++pos;
        }
    }
}

// ---------------------------------------------------------------- feature gather * tanh
// One wave per output row; each lane moves one float4 (128 floats = 32 lanes * 4).
__global__ void k_gather(const float* __restrict__ x, const float* __restrict__ score,
                         const int* __restrict__ perm, const int* __restrict__ perm_com,
                         float* out_dis, float* out_com) {
    int gidx = blockIdx.x * blockDim.x + threadIdx.x;
    int row  = gidx >> 5;
    int lane = gidx & 31;
    int node;
    float* orow;
    if (row < BGRAPH * KKEEP) {
        node = perm[row];
        orow = out_dis + (size_t)row * DIM;
    } else {
        int r2 = row - BGRAPH * KKEEP;
        node = perm_com[r2];
        orow = out_com + (size_t)r2 * DIM;
    }
    float tt = tanhf(score[node]);
    float4 v = ((const float4*)(x + (size_t)node * DIM))[lane];
    v.x *= tt; v.y *= tt; v.z *= tt; v.w *= tt;
    ((float4*)orow)[lane] = v;
}

// ---------------------------------------------------------------- launcher
extern "C" void kernel_launch(void* const* d_in, const int* in_sizes, int n_in,
                              void* d_out, int out_size, void* d_ws, size_t ws_size,
                              hipStream_t stream) {
    (void)in_sizes; (void)n_in; (void)out_size; (void)ws_size;
    const float* x    = (const float*)d_in[0];
    const float* W    = (const float*)d_in[1];
    const float* bias = (const float*)d_in[2];
    const int*   src  = (const int*)d_in[3];
    const int*   dst  = (const int*)d_in[4];

    // workspace layout
    unsigned* deg_out = (unsigned*)d_ws;
    unsigned* deg_in  = deg_out + NTOT;
    float*    h       = (float*)(deg_in + NTOT);
    float*    agg     = h + NTOT;
    float*    score   = agg + NTOT;
    float*    red     = score + NTOT;            // red[0]=max, red[1]=sumexp

    // output layout (tuple flattened in return order)
    float* out        = (float*)d_out;
    float* out_dis    = out;                                        // [B*K, D]
    float* out_com    = out + (size_t)BGRAPH * KKEEP * DIM;         // [B*(N-K), D]
    int*   out_perm   = (int*)(out + 2 * (size_t)BGRAPH * KKEEP * DIM);
    int*   out_pcom   = out_perm + BGRAPH * KKEEP;
    float* out_soft   = (float*)(out_pcom + BGRAPH * (NNODE - KKEEP));

    const dim3 blk(256);
    k_init  <<<(NTOT + 255) / 256, blk, 0, stream>>>(deg_out, deg_in, agg, red);
    k_deg   <<<(NEDGE + 255) / 256, blk, 0, stream>>>(src, dst, deg_out, deg_in);
    // NTOT/16 waves, 8 waves per block
    k_h_wmma<<<NTOT / 16 / 8, blk, 0, stream>>>(x, W, h);
    k_edge  <<<(NEDGE + 255) / 256, blk, 0, stream>>>(src, dst, h, deg_out, agg);
    k_score <<<NTOT / 256, blk, 0, stream>>>(agg, deg_in, bias, score, red);
    k_sumexp<<<NTOT / 256, blk, 0, stream>>>(score, red);
    k_soft  <<<NTOT / 256, blk, 0, stream>>>(score, red, out_soft);
    k_topk  <<<BGRAPH, 1024, 0, stream>>>(score, out_perm, out_pcom);
    // one wave per row, B*N rows total
    k_gather<<<(size_t)BGRAPH * NNODE * 32 / 256, blk, 0, stream>>>(
        x, score, out_perm, out_pcom, out_dis, out_com);
}